// TemporalAttention_249108103968
// MI455X (gfx1250) — compile-verified
//
#include <hip/hip_runtime.h>
#include <math.h>

typedef __attribute__((ext_vector_type(2))) float v2f;
typedef __attribute__((ext_vector_type(8))) float v8f;

#define NN   512
#define BB   4
#define BN   2048        // B*N
#define HID  64
#define TENC 8
#define TDEC 4
#define EE   8192
#define F1   129         // 65 + 64
#define F2   516         // F1 * B
#define NM   5           // NUM_MATS
#define KZ   645         // F1*NM
#define KZP  648         // padded to mult of 4
#define CAP  128         // max edges per destination row (mean is 16)

// ---------------------------------------------------------------------------
// Generic wave32 WMMA f32 GEMM: C(M x N) = act(A(M x Kp) @ Bm(Kp x Np) + bias)
// One 32-thread block computes a 16 x (16*NT) strip: A fragment reused NT x.
// M, Np multiples of 16; Kp multiple of 4. act: 0=none 1=sigmoid 2=tanh 3=relu
// ---------------------------------------------------------------------------
template <int NT>
__global__ __launch_bounds__(32) void gemm_wmma_f32(
    const float* __restrict__ A, int lda,
    const float* __restrict__ Bm, int ldb,
    const float* __restrict__ bias,
    float* __restrict__ C, int ldc,
    int Ncols, int Kp, int act)
{
  const int tileM  = blockIdx.x;
  const int tileN0 = blockIdx.y * NT;
  const int lane   = threadIdx.x;
  const int hl     = lane >> 4;     // 0/1
  const int l16    = lane & 15;

  const int rowA = tileM * 16 + l16;

  v8f acc[NT];
  #pragma unroll
  for (int j = 0; j < NT; ++j) acc[j] = (v8f){};

  for (int k0 = 0; k0 < Kp; k0 += 4) {
    const int ka = k0 + hl * 2;     // lanes 0-15 -> K {k0,k0+1}; 16-31 -> {k0+2,k0+3}
    v2f a;
    a[0] = A[(size_t)rowA * lda + ka];
    a[1] = A[(size_t)rowA * lda + ka + 1];
    #pragma unroll
    for (int j = 0; j < NT; ++j) {
      const int colB = (tileN0 + j) * 16 + l16;
      v2f b;
      b[0] = Bm[(size_t)ka * ldb + colB];
      b[1] = Bm[(size_t)(ka + 1) * ldb + colB];
      acc[j] = __builtin_amdgcn_wmma_f32_16x16x4_f32(false, a, false, b,
                                                     (short)0, acc[j], false, false);
    }
  }

  #pragma unroll
  for (int j = 0; j < NT; ++j) {
    const int c = (tileN0 + j) * 16 + l16;
    if (c < Ncols) {
      const float bv = bias[c];
      #pragma unroll
      for (int v = 0; v < 8; ++v) {
        const int r = tileM * 16 + v + hl * 8;   // C layout: VGPR v -> rows v / v+8
        float val = acc[j][v] + bv;
        if (act == 1)      val = 1.0f / (1.0f + expf(-val));
        else if (act == 2) val = tanhf(val);
        else if (act == 3) val = fmaxf(val, 0.0f);
        C[(size_t)r * ldc + c] = val;
      }
    }
  }
}

// ---------------------------------------------------------------------------
// CSR build (once per launch, per support): one thread per destination row
// scans edges in ascending id order -> deterministic, matches segment order.
// ---------------------------------------------------------------------------
__global__ void build_csr(const int* __restrict__ rows,
                          int* __restrict__ cnt, int* __restrict__ lst) {
  const int n = blockIdx.x * blockDim.x + threadIdx.x;
  if (n >= NN) return;
  int c = 0;
  for (int e = 0; e < EE; ++e) {
    if (rows[e] == n) {
      if (c < CAP) lst[n * CAP + c] = e;
      ++c;
    }
  }
  cnt[n] = (c > CAP) ? CAP : c;
}

// ---------------------------------------------------------------------------
// SpMM via per-row edge list: Y[n,:] = sum_e vals[e] * X[cols[e],:]  (F=516)
// ---------------------------------------------------------------------------
__global__ __launch_bounds__(256) void spmm_csr(
    const int* __restrict__ cnt, const int* __restrict__ lst,
    const int* __restrict__ cols, const float* __restrict__ vals,
    const float* __restrict__ X, float* __restrict__ Y)
{
  const int n = blockIdx.x;
  const int tid = threadIdx.x;
  const int c = cnt[n];
  const int f0 = tid, f1 = tid + 256, f2 = tid + 512;
  float a0 = 0.f, a1 = 0.f, a2 = 0.f;
  for (int i = 0; i < c; ++i) {
    const int e = lst[n * CAP + i];
    const float v = vals[e];
    const float* xr = X + (size_t)cols[e] * F2;
    a0 += v * xr[f0];
    if (f1 < F2) a1 += v * xr[f1];
    if (f2 < F2) a2 += v * xr[f2];
  }
  Y[(size_t)n * F2 + f0] = a0;
  if (f1 < F2) Y[(size_t)n * F2 + f1] = a1;
  if (f2 < F2) Y[(size_t)n * F2 + f2] = a2;
}

// ---------------------------------------------------------------------------
// Temporal attention: s[t]=sum_{c,k} v[k]*tanh(hf[c,t]+y[k]); softmax over t;
// attn[c]=sum_t a[t]*hidden4[c*8+t].  One 256-thread block per (b,n).
// ---------------------------------------------------------------------------
__global__ __launch_bounds__(256) void attention_kernel(
    const float* __restrict__ hfT,      // (BN*8, 64): hf[bn,c,t] at ((bn*8+t)*64+c)
    const float* __restrict__ hidden4,  // (BN, 512)
    const float* __restrict__ ybuf,     // (BN, 64)
    const float* __restrict__ vvec,     // (64)
    float* __restrict__ attn)           // (BN, 64)
{
  __shared__ float sy[64], sv[64], shf[8][64], red[256], s_t[8];
  const int bn = blockIdx.x;
  const int tid = threadIdx.x;
  if (tid < 64) { sy[tid] = ybuf[(size_t)bn * 64 + tid]; sv[tid] = vvec[tid]; }
  for (int i = tid; i < 512; i += 256) {
    const int t = i >> 6, c = i & 63;
    shf[t][c] = hfT[((size_t)bn * 8 + t) * 64 + c];
  }
  __syncthreads();

  const int c  = tid >> 2;
  const int k0 = (tid & 3) * 16;
  float accs[8];
  #pragma unroll
  for (int t = 0; t < 8; ++t) {
    const float h = shf[t][c];
    float a = 0.f;
    for (int k = k0; k < k0 + 16; ++k) a += sv[k] * tanhf(h + sy[k]);
    accs[t] = a;
  }
  for (int t = 0; t < 8; ++t) {
    red[tid] = accs[t];
    __syncthreads();
    for (int off = 128; off > 0; off >>= 1) {
      if (tid < off) red[tid] += red[tid + off];
      __syncthreads();
    }
    if (tid == 0) s_t[t] = red[0];
    __syncthreads();
  }
  float m = s_t[0];
  for (int t = 1; t < 8; ++t) m = fmaxf(m, s_t[t]);
  float w[8], den = 0.f;
  for (int t = 0; t < 8; ++t) { w[t] = expf(s_t[t] - m); den += w[t]; }
  if (tid < 64) {
    float o = 0.f;
    for (int t = 0; t < 8; ++t)
      o += (w[t] / den) * hidden4[(size_t)bn * 512 + tid * 8 + t];
    attn[(size_t)bn * 64 + tid] = o;
  }
}

// ------------------------- elementwise / pack kernels ----------------------
__global__ void build_hidden4(const float* __restrict__ enc, float* __restrict__ h4) {
  int i = blockIdx.x * blockDim.x + threadIdx.x;
  if (i >= BN * 512) return;
  const int bn = i >> 9, idx = i & 511;
  const int b = bn >> 9, n = bn & 511;
  const int te = idx >> 6, h = idx & 63;
  h4[i] = enc[(((size_t)(b * TENC + te)) * NN + n) * HID + h];
}
__global__ void pack_xht(const float* __restrict__ h4, float* __restrict__ X) {
  int i = blockIdx.x * blockDim.x + threadIdx.x;
  if (i >= BN * 8 * 64) return;
  const int r = i >> 6, c = i & 63;
  const int bn = r >> 3, t = r & 7;
  X[i] = h4[(size_t)bn * 512 + c * 8 + t];
}
__global__ void pack_pad(const float* __restrict__ W, float* __restrict__ Wp,
                         int K, int Nc, int Kp, int Np) {
  int i = blockIdx.x * blockDim.x + threadIdx.x;
  if (i >= Kp * Np) return;
  const int k = i / Np, n = i % Np;
  Wp[i] = (k < K && n < Nc) ? W[(size_t)k * Nc + n] : 0.f;
}
__global__ void pack_transpose(const float* __restrict__ W, float* __restrict__ Wt,
                               int R, int Cc) {  // W (R,Cc) -> Wt (Cc,R)
  int i = blockIdx.x * blockDim.x + threadIdx.x;
  if (i >= R * Cc) return;
  const int c = i / R, r = i % R;
  Wt[(size_t)c * R + r] = W[(size_t)r * Cc + c];
}
__global__ void build_xin(const float* __restrict__ dec_t,
                          const float* __restrict__ attn, float* __restrict__ Xin) {
  int i = blockIdx.x * blockDim.x + threadIdx.x;
  if (i >= BN * 68) return;
  const int bn = i / 68, j = i % 68;
  float v = 0.f;
  if (j == 0)      v = dec_t[bn];
  else if (j < 65) v = attn[(size_t)bn * 64 + (j - 1)];
  Xin[i] = v;
}
__global__ void pack_x0(const float* __restrict__ xb, const float* __restrict__ st,
                        float* __restrict__ m0) {
  int i = blockIdx.x * blockDim.x + threadIdx.x;
  if (i >= NN * F2) return;
  const int n = i / F2, col = i % F2;
  const int f = col >> 2, b = col & 3;
  const int bn = b * NN + n;
  m0[i] = (f < 65) ? xb[(size_t)bn * 65 + f] : st[(size_t)bn * 64 + (f - 65)];
}
__global__ void pack_z(const float* __restrict__ mats, float* __restrict__ Z) {
  int i = blockIdx.x * blockDim.x + threadIdx.x;
  if (i >= BN * KZP) return;
  const int bn = i / KZP, col = i % KZP;
  float v = 0.f;
  if (col < KZ) {
    const int f = col / NM, m = col % NM;
    const int b = bn >> 9, n = bn & 511;
    v = mats[((size_t)m * NN + n) * F2 + f * 4 + b];
  }
  Z[i] = v;
}
__global__ void mul_r_state(const float* __restrict__ g, const float* __restrict__ hs,
                            float* __restrict__ rs) {
  int i = blockIdx.x * blockDim.x + threadIdx.x;
  if (i >= BN * 64) return;
  const int bn = i >> 6, h = i & 63;
  rs[i] = g[(size_t)bn * 128 + h] * hs[i];
}
__global__ void make_new_state(const float* __restrict__ g, const float* __restrict__ hs,
                               const float* __restrict__ c, float* __restrict__ ns) {
  int i = blockIdx.x * blockDim.x + threadIdx.x;
  if (i >= BN * 64) return;
  const int bn = i >> 6, h = i & 63;
  const float u = g[(size_t)bn * 128 + 64 + h];
  ns[i] = u * hs[i] + (1.f - u) * c[i];
}
__global__ void concat2(const float* __restrict__ a, const float* __restrict__ b,
                        float* __restrict__ o) {
  int i = blockIdx.x * blockDim.x + threadIdx.x;
  if (i >= BN * 128) return;
  const int bn = i >> 7, j = i & 127;
  o[i] = (j < 64) ? a[(size_t)bn * 64 + j] : b[(size_t)bn * 64 + j - 64];
}

static inline int cdiv(int a, int b) { return (a + b - 1) / b; }

extern "C" void kernel_launch(void* const* d_in, const int* in_sizes, int n_in,
                              void* d_out, int out_size, void* d_ws, size_t ws_size,
                              hipStream_t stream) {
  const float* dec   = (const float*)d_in[0];
  const float* hst   = (const float*)d_in[2];   // initial hidden_state (B,N,H)
  const float* enc   = (const float*)d_in[3];
  const int*   rows  = (const int*)  d_in[4];
  const int*   cols  = (const int*)  d_in[5];
  const float* vals  = (const float*)d_in[6];
  const float* Wg    = (const float*)d_in[7];
  const float* bg    = (const float*)d_in[8];
  const float* Wc    = (const float*)d_in[9];
  const float* bc    = (const float*)d_in[10];
  const float* p1W   = (const float*)d_in[11];
  const float* p1b   = (const float*)d_in[12];
  const float* convW = (const float*)d_in[13];
  const float* convb = (const float*)d_in[14];
  const float* vvec  = (const float*)d_in[15];
  const float* attn0 = (const float*)d_in[16];
  const float* qW    = (const float*)d_in[17];
  const float* qb    = (const float*)d_in[18];
  const float* inW   = (const float*)d_in[19];
  const float* inb   = (const float*)d_in[20];
  const float* outW  = (const float*)d_in[21];
  const float* outb  = (const float*)d_in[22];
  float* out = (float*)d_out;

  float* ws = (float*)d_ws;
  size_t o = 0;
  float* hidden4 = ws + o; o += (size_t)BN * 512;
  float* hfT     = ws + o; o += (size_t)BN * 8 * 64;
  float* attnb   = ws + o; o += (size_t)BN * 64;
  float* Xin     = ws + o; o += (size_t)BN * 68;
  float* xbuf    = ws + o; o += (size_t)BN * 65;
  float* mats    = ws + o; o += (size_t)NM * NN * F2;
  float* Z       = ws + o; o += (size_t)BN * KZP;   // also aliased as XhT in precompute
  float* gbuf    = ws + o; o += (size_t)BN * 128;
  float* rsb     = ws + o; o += (size_t)BN * 64;
  float* cbuf    = ws + o; o += (size_t)BN * 64;
  float* nstate  = ws + o; o += (size_t)BN * 64;
  float* obuf    = ws + o; o += (size_t)BN * 64;
  float* ybuf    = ws + o; o += (size_t)BN * 64;
  float* catb    = ws + o; o += (size_t)BN * 128;
  float* Wgp     = ws + o; o += (size_t)KZP * 128;
  float* Wcp     = ws + o; o += (size_t)KZP * 64;
  float* inWp    = ws + o; o += (size_t)68 * 80;
  float* convWT  = ws + o; o += (size_t)64 * 64;
  int* iws = (int*)(ws + o);
  int* cnt0 = iws;                 // 512
  int* cnt1 = iws + NN;            // 512
  int* lst0 = iws + 2 * NN;        // 512*CAP
  int* lst1 = lst0 + NN * CAP;     // 512*CAP
  (void)ws_size; (void)in_sizes; (void)n_in; (void)out_size;

  const int BS = 256;
  #define EW(kern, cnt, ...) kern<<<cdiv((cnt), BS), BS, 0, stream>>>(__VA_ARGS__)

  // ---- once-per-launch precompute ----
  hipMemcpyAsync(attnb, attn0, (size_t)BN * 64 * sizeof(float),
                 hipMemcpyDeviceToDevice, stream);
  const int* r0 = rows, *r1 = rows + EE;
  const int* c0 = cols, *c1 = cols + EE;
  const float* v0 = vals, *v1 = vals + EE;
  build_csr<<<cdiv(NN, BS), BS, 0, stream>>>(r0, cnt0, lst0);
  build_csr<<<cdiv(NN, BS), BS, 0, stream>>>(r1, cnt1, lst1);

  EW(build_hidden4, BN * 512, enc, hidden4);
  float* XhT = Z;  // alias: used before Z is needed
  EW(pack_xht, BN * 8 * 64, hidden4, XhT);
  EW(pack_transpose, 64 * 64, convW, convWT, 64, 64);
  EW(pack_pad, KZP * 128, Wg, Wgp, KZ, 128, KZP, 128);
  EW(pack_pad, KZP * 64,  Wc, Wcp, KZ, 64,  KZP, 64);
  EW(pack_pad, 68 * 80,   inW, inWp, 65, 65, 68, 80);
  // hf = conv over hidden: (BN*8 x 64) = XhT @ convW^T + convb
  gemm_wmma_f32<4><<<dim3(BN * 8 / 16, 1), 32, 0, stream>>>(
      XhT, 64, convWT, 64, convb, hfT, 64, 64, 64, 0);

  float* m0 = mats;
  float* m1 = mats + (size_t)1 * NN * F2;
  float* m2 = mats + (size_t)2 * NN * F2;
  float* m3 = mats + (size_t)3 * NN * F2;
  float* m4 = mats + (size_t)4 * NN * F2;

  for (int t = 0; t < TDEC; ++t) {
    // x = [inp_t | attn] @ inW + inb
    EW(build_xin, BN * 68, dec + (size_t)t * BN, attnb, Xin);
    gemm_wmma_f32<5><<<dim3(BN / 16, 1), 32, 0, stream>>>(
        Xin, 68, inWp, 80, inb, xbuf, 65, 65, 68, 0);

    // dconv #1 (gates): state = initial hidden_state
    EW(pack_x0, NN * F2, xbuf, hst, m0);
    spmm_csr<<<NN, 256, 0, stream>>>(cnt0, lst0, c0, v0, m0, m1);
    spmm_csr<<<NN, 256, 0, stream>>>(cnt0, lst0, c0, v0, m1, m2);
    spmm_csr<<<NN, 256, 0, stream>>>(cnt1, lst1, c1, v1, m0, m3);
    spmm_csr<<<NN, 256, 0, stream>>>(cnt1, lst1, c1, v1, m3, m4);
    EW(pack_z, BN * KZP, mats, Z);
    gemm_wmma_f32<4><<<dim3(BN / 16, 2), 32, 0, stream>>>(
        Z, KZP, Wgp, 128, bg, gbuf, 128, 128, KZP, 1 /*sigmoid*/);

    // dconv #2 (candidate): state = r * hidden_state
    EW(mul_r_state, BN * 64, gbuf, hst, rsb);
    EW(pack_x0, NN * F2, xbuf, rsb, m0);
    spmm_csr<<<NN, 256, 0, stream>>>(cnt0, lst0, c0, v0, m0, m1);
    spmm_csr<<<NN, 256, 0, stream>>>(cnt0, lst0, c0, v0, m1, m2);
    spmm_csr<<<NN, 256, 0, stream>>>(cnt1, lst1, c1, v1, m0, m3);
    spmm_csr<<<NN, 256, 0, stream>>>(cnt1, lst1, c1, v1, m3, m4);
    EW(pack_z, BN * KZP, mats, Z);
    gemm_wmma_f32<4><<<dim3(BN / 16, 1), 32, 0, stream>>>(
        Z, KZP, Wcp, 64, bc, cbuf, 64, 64, KZP, 2 /*tanh*/);

    // new_state, projection, query
    EW(make_new_state, BN * 64, gbuf, hst, cbuf, nstate);
    gemm_wmma_f32<4><<<dim3(BN / 16, 1), 32, 0, stream>>>(
        nstate, 64, p1W, 64, p1b, obuf, 64, 64, 64, 3 /*relu*/);
    gemm_wmma_f32<4><<<dim3(BN / 16, 1), 32, 0, stream>>>(
        nstate, 64, qW, 64, qb, ybuf, 64, 64, 64, 0);

    // temporal attention -> attnb (carried to next step)
    attention_kernel<<<BN, 256, 0, stream>>>(hfT, hidden4, ybuf, vvec, attnb);

    // o = [out | attn2] @ outW + outb  -> outputs[t]
    EW(concat2, BN * 128, obuf, attnb, catb);
    gemm_wmma_f32<4><<<dim3(BN / 16, 1), 32, 0, stream>>>(
        catb, 128, outW, 64, outb, out + (size_t)t * BN * 64, 64, 64, 128, 0);
  }

  // final_state = last new_state
  hipMemcpyAsync(out + (size_t)TDEC * BN * 64, nstate,
                 (size_t)BN * 64 * sizeof(float), hipMemcpyDeviceToDevice, stream);
  #undef EW
}